// Render_49134425866981
// MI455X (gfx1250) — compile-verified
//
#include <hip/hip_runtime.h>
#include <stdint.h>

// CDNA5 / gfx1250 rasterizer, round 2.
// - V_WMMA_F32_16X16X4_F32 evaluates 8 affine quantities x 2 faces for 16
//   pixels per instruction (A = face coeffs, B = (X,Y,1,0) pixel rows).
// - Coefficients + bboxes are staged into double-buffered LDS with
//   GLOBAL_LOAD_ASYNC_TO_LDS_B128 (ASYNCcnt), overlapped with compute.
// - Per-lane running argmax over its parity stream; single shfl_xor(16)
//   merge at the end (tie: lower face id wins == reference "first max").

typedef __attribute__((ext_vector_type(2))) float v2f;
typedef __attribute__((ext_vector_type(8))) float v8f;

#define H_IMG 384
#define W_IMG 384
#define NFACES 2048
#define SEGS (W_IMG / 16)    // 24 sixteen-pixel segments per row
#define NPAIRS (NFACES / 2)  // 1024 face pairs
#define CH 32                // face pairs per LDS chunk
#define NCHUNK (NPAIRS / CH) // 32 chunks

// 16-byte async copy: global -> LDS. Low 32 bits of a generic pointer to
// __shared__ memory are the LDS byte address on AMDGPU.
__device__ __forceinline__ void async_cp16(void* lds_dst, const void* gsrc) {
  uint32_t l = (uint32_t)(uintptr_t)lds_dst;
  asm volatile("global_load_async_to_lds_b128 %0, %1, off"
               :
               : "v"(l), "v"(gsrc)
               : "memory");
}

__device__ __forceinline__ void wait_async0() {
  asm volatile("s_wait_asynccnt 0x0" ::: "memory");
}

// ---------------------------------------------------------------------------
// Setup: per-face affine coefficients + bbox (same as round 1).
// coef[f*8 + q] = float4(a, b, c, 0): quantity_q(X,Y) = a*X + b*Y + c.
// q: 0=t1 1=t2 2=t3 3=c1 4=c2 5=c3 6=z 7=pad
// ---------------------------------------------------------------------------
__global__ void raster_setup(const float* __restrict__ verts,
                             const int* __restrict__ faces,
                             float4* __restrict__ coef,
                             float4* __restrict__ bbox) {
  int f = blockIdx.x * blockDim.x + threadIdx.x;
  if (f >= NFACES) return;
  int ia = faces[f * 3 + 0], ib = faces[f * 3 + 1], ic = faces[f * 3 + 2];
  float A0 = verts[ia * 3 + 0], A1 = verts[ia * 3 + 1], Az = verts[ia * 3 + 2];
  float B0 = verts[ib * 3 + 0], B1 = verts[ib * 3 + 1], Bz = verts[ib * 3 + 2];
  float C0 = verts[ic * 3 + 0], C1 = verts[ic * 3 + 1];
  // C.z intentionally unused: reference computes z = c1*Az + c2*Bz + c3*Bz.

  float t1a = A1 - B1, t1b = B0 - A0, t1c = A0 * B1 - B0 * A1;
  float t2a = B1 - C1, t2b = C0 - B0, t2c = B0 * C1 - C0 * B1;
  float t3a = C1 - A1, t3b = A0 - C0, t3c = C0 * A1 - A0 * C1;

  float area = (B0 - A0) * (C1 - A1) - (B1 - A1) * (C0 - A0);
  float ra = 1.0f / area;

  float c1a = (B1 - C1) * ra, c1b = -(B0 - C0) * ra;
  float c1c = (-C0 * (B1 - C1) + C1 * (B0 - C0)) * ra;
  float c2a = (C1 - A1) * ra, c2b = -(C0 - A0) * ra;
  float c2c = (-A0 * (C1 - A1) + A1 * (C0 - A0)) * ra;
  float c3a = -(B1 - A1) * ra, c3b = (B0 - A0) * ra;
  float c3c = (A0 * (B1 - A1) - A1 * (B0 - A0)) * ra;

  float za = c1a * Az + (c2a + c3a) * Bz;
  float zb = c1b * Az + (c2b + c3b) * Bz;
  float zc = c1c * Az + (c2c + c3c) * Bz;

  float4* cp = coef + (size_t)f * 8;
  cp[0] = make_float4(t1a, t1b, t1c, 0.f);
  cp[1] = make_float4(t2a, t2b, t2c, 0.f);
  cp[2] = make_float4(t3a, t3b, t3c, 0.f);
  cp[3] = make_float4(c1a, c1b, c1c, 0.f);
  cp[4] = make_float4(c2a, c2b, c2c, 0.f);
  cp[5] = make_float4(c3a, c3b, c3c, 0.f);
  cp[6] = make_float4(za, zb, zc, 0.f);
  cp[7] = make_float4(0.f, 0.f, 0.f, 0.f);

  float xmn = fminf(A0, fminf(B0, C0)), xmx = fmaxf(A0, fmaxf(B0, C0));
  float ymn = fminf(A1, fminf(B1, C1)), ymx = fmaxf(A1, fmaxf(B1, C1));
  float x_min = fmaxf(truncf(xmn), 0.f);
  float x_max = fminf(truncf(xmx), (float)(W_IMG - 1));
  float y_min = fmaxf(truncf(ymn), 0.f);
  float y_max = fminf(truncf(ymx), (float)(H_IMG - 1));
  bool valid = (x_min < x_max) && (y_min < y_max);
  if (!valid) { x_min = 1.f; x_max = 0.f; y_min = 1.f; y_max = 0.f; }
  bbox[f] = make_float4(x_min, x_max, y_min, y_max);
}

// ---------------------------------------------------------------------------
// Main: 8 waves/block, each wave owns a 16-pixel row segment. All waves share
// the face stream staged in LDS. Lanes 0-15 carry even faces of a pair,
// lanes 16-31 odd faces.
// ---------------------------------------------------------------------------
__global__ __launch_bounds__(256) void raster_main(
    const float4* __restrict__ coef, const float4* __restrict__ bbox,
    float* __restrict__ out) {
  __shared__ float4 coefS[2][CH * 16];  // 2 x 8 KB
  __shared__ float4 bboxS[2][CH * 2];   // 2 x 1 KB

  int t = threadIdx.x;
  int tid = blockIdx.x * 256 + t;
  int wave = tid >> 5;
  int lane = t & 31;
  int row = wave / SEGS;        // X = Xg = first image axis
  int j0 = (wave % SEGS) << 4;  // Y = Yg = second image axis
  int hi = lane >> 4;           // 0: face f0 of pair, 1: face f1
  int p = lane & 15;

  float pX = (float)row;
  float pY = (float)(j0 + p);

  // B matrix (4x16): lanes 0-15 -> rows K0=X, K1=Y; lanes 16-31 -> K2=1, K3=0.
  v2f Bm;
  Bm.x = hi ? 1.0f : pX;
  Bm.y = hi ? 0.0f : pY;

  float best_z = -1000.0f, best_f = 0.0f;
  float bc1 = 0.f, bc2 = 0.f, bc3 = 0.f;

  // Stage chunk 0.
  async_cp16(&coefS[0][t], coef + t);
  async_cp16(&coefS[0][t + 256], coef + t + 256);
  if (t < CH * 2) async_cp16(&bboxS[0][t], bbox + t);
  wait_async0();
  __syncthreads();

  for (int c = 0; c < NCHUNK; ++c) {
    int b = c & 1;
    if (c + 1 < NCHUNK) {  // overlap: async-stage next chunk into other buffer
      const float4* gs = coef + (size_t)(c + 1) * (CH * 16);
      async_cp16(&coefS[b ^ 1][t], gs + t);
      async_cp16(&coefS[b ^ 1][t + 256], gs + t + 256);
      if (t < CH * 2)
        async_cp16(&bboxS[b ^ 1][t], bbox + (size_t)(c + 1) * (CH * 2) + t);
    }
    if (c + 2 < NCHUNK)
      __builtin_prefetch(coef + (size_t)(c + 2) * (CH * 16), 0, 3);

#pragma unroll 4
    for (int k = 0; k < CH; ++k) {
      // A matrix row M = quantity (lane&15), lane half selects (a,b) vs (c,0).
      const float* ap = (const float*)&coefS[b][k * 16 + p] + (hi ? 2 : 0);
      v2f Am = *(const v2f*)ap;                  // ds_load_b64
      float4 bb = bboxS[b][k * 2 + hi];          // ds_load_b128

      v8f acc = {0.f, 0.f, 0.f, 0.f, 0.f, 0.f, 0.f, 0.f};
      acc = __builtin_amdgcn_wmma_f32_16x16x4_f32(
          /*neg_a=*/false, Am, /*neg_b=*/false, Bm,
          /*c_mod=*/(short)0, acc, /*reuse_a=*/false, /*reuse_b=*/false);

      // Bitwise & (no short-circuit) -> straight-line cndmask, EXEC untouched.
      int ins = (int)(acc[0] * acc[1] >= 0.f) & (int)(acc[1] * acc[2] >= 0.f) &
                (int)(pX >= bb.x) & (int)(pX <= bb.y) &
                (int)(pY >= bb.z) & (int)(pY <= bb.w);
      int upd = ins & (int)(acc[6] > best_z);  // strict >: earliest face wins
      float fidv = (float)(((c * CH + k) << 1) | hi);
      best_z = upd ? acc[6] : best_z;
      best_f = upd ? fidv : best_f;
      bc1 = upd ? acc[3] : bc1;
      bc2 = upd ? acc[4] : bc2;
      bc3 = upd ? acc[5] : bc3;
    }

    wait_async0();   // next chunk's LDS writes landed
    __syncthreads(); // and everyone is done reading this chunk's buffer
  }

  // Merge even/odd face streams (partner lane = lane^16, same pixel).
  // Tie on z -> lower face id wins == reference's first-max semantics.
  float zo = __shfl_xor(best_z, 16, 32);
  float fo = __shfl_xor(best_f, 16, 32);
  float c1o = __shfl_xor(bc1, 16, 32);
  float c2o = __shfl_xor(bc2, 16, 32);
  float c3o = __shfl_xor(bc3, 16, 32);
  int take = (int)(zo > best_z) | ((int)(zo == best_z) & (int)(fo < best_f));
  float rz = take ? zo : best_z;
  float rf = take ? fo : best_f;
  float r1 = take ? c1o : bc1;
  float r2 = take ? c2o : bc2;
  float r3 = take ? c3o : bc3;

  if (!hi) {
    size_t o = ((size_t)row * W_IMG + j0 + p) * 5;
    out[o + 0] = rf;
    out[o + 1] = r1;
    out[o + 2] = r2;
    out[o + 3] = r3;
    out[o + 4] = rz;
  }
}

extern "C" void kernel_launch(void* const* d_in, const int* in_sizes, int n_in,
                              void* d_out, int out_size, void* d_ws,
                              size_t ws_size, hipStream_t stream) {
  const float* verts = (const float*)d_in[0];  // (1, 4096, 3) f32
  const int* faces = (const int*)d_in[1];      // (1, 2048, 3) i32
  float* out = (float*)d_out;                  // (1,384,384,1,5) f32

  // Workspace: coef table (2048*8 float4 = 256KB) then bbox (2048 float4).
  float4* coef = (float4*)d_ws;
  float4* bbox = coef + (size_t)NFACES * 8;

  raster_setup<<<(NFACES + 255) / 256, 256, 0, stream>>>(verts, faces, coef,
                                                         bbox);
  int waves = H_IMG * SEGS;  // 9216 waves, 8 waves per 256-thread block
  raster_main<<<waves / 8, 256, 0, stream>>>(coef, bbox, out);
}